// ArcNetAhmed_61589831025165
// MI455X (gfx1250) — compile-verified
//
#include <hip/hip_runtime.h>
#include <math.h>

// Problem constants from the reference
static constexpr int B = 2048;
static constexpr int C = 50000;   // divisible by 16 (50000 = 16*3125)
static constexpr int D = 128;
static constexpr float SCALE_S = 20.0f;
static constexpr float MARGIN  = 0.1f;
static constexpr float EPS     = 1e-7f;

typedef __attribute__((ext_vector_type(2))) float v2f;
typedef __attribute__((ext_vector_type(8))) float v8f;

// ---------------------------------------------------------------------------
// Row L2 normalize: out[i,:] = in[i,:] / max(||in[i,:]||, 1e-12)
// One 128-thread block per row; D == 128 so thread t owns column t.
// ---------------------------------------------------------------------------
__global__ __launch_bounds__(128) void l2norm_rows(const float* __restrict__ in,
                                                   float* __restrict__ out) {
  const int i = blockIdx.x;
  const int t = threadIdx.x;
  const float v = in[(size_t)i * D + t];
  __shared__ float red[128];
  red[t] = v * v;
  __syncthreads();
#pragma unroll
  for (int s = 64; s > 0; s >>= 1) {
    if (t < s) red[t] += red[t + s];
    __syncthreads();
  }
  __shared__ float scale;
  if (t == 0) scale = 1.0f / fmaxf(sqrtf(red[0]), 1e-12f);
  __syncthreads();
  out[(size_t)i * D + t] = v * scale;
}

// ---------------------------------------------------------------------------
// GEMM: wf[B,C] = e[B,D] @ Wn[C,D]^T + bias[C]   via V_WMMA_F32_16X16X4_F32
// Block = 256 threads = 8 waves; block tile = 32 rows x 64 cols;
// each wave computes one 16x16 output tile, looping K in steps of 4.
//
// Operand packing per CDNA5 ISA (wave32):
//   A 16x4 f32:  lane<16 -> {K=k0,k0+1} of row M=lane; lane>=16 -> {K=k0+2,k0+3}
//   B 4x16 f32:  lane<16 -> rows {k0,k0+1} at col N=lane&15; lane>=16 -> {k0+2,k0+3}
//   C/D 16x16:   acc[r] = C[M = r + 8*(lane>=16)][N = lane&15]
// ---------------------------------------------------------------------------
__global__ __launch_bounds__(256) void gemm_wmma_f32(const float* __restrict__ e,
                                                     const float* __restrict__ Wn,
                                                     const float* __restrict__ bias,
                                                     float* __restrict__ wf) {
  const int lane = threadIdx.x & 31;
  const int wave = threadIdx.x >> 5;          // 0..7
  const int tr   = wave >> 2;                 // 0..1  (row tile within block)
  const int tc   = wave & 3;                  // 0..3  (col tile within block)
  const int row0 = blockIdx.y * 32 + tr * 16;
  const int col0 = blockIdx.x * 64 + tc * 16;
  if (col0 >= C) return;                      // wave-uniform: EXEC stays all-ones

  const int half = lane >> 4;                 // 0 or 1
  const int lm   = lane & 15;

  const float* __restrict__ arow = e  + (size_t)(row0 + lm) * D + 2 * half;
  const float* __restrict__ brow = Wn + (size_t)(col0 + lm) * D + 2 * half;

  v8f acc = {};
#pragma unroll
  for (int kk = 0; kk < D; kk += 4) {
    v2f a, b;
    a.x = arow[kk];
    a.y = arow[kk + 1];
    b.x = brow[kk];
    b.y = brow[kk + 1];
    // 8 args: (neg_a, A, neg_b, B, c_mod, C, reuse_a, reuse_b)
    acc = __builtin_amdgcn_wmma_f32_16x16x4_f32(false, a, false, b,
                                                (short)0, acc, false, false);
  }

  const float bc = bias[col0 + lm];
#pragma unroll
  for (int r = 0; r < 8; ++r) {
    const int gr = row0 + r + 8 * half;
    wf[(size_t)gr * C + col0 + lm] = acc[r] + bc;
  }
}

// ---------------------------------------------------------------------------
// Zero the loss accumulator slot.
// ---------------------------------------------------------------------------
__global__ void zero_loss(float* loss) { *loss = 0.0f; }

// ---------------------------------------------------------------------------
// Per-row pass: online softmax stats for prediction (raw wf) and for the
// margin-modified, S-scaled loss logits. One 256-thread block per row.
//   stats[2i]   = rowmax(wf[i,:])
//   stats[2i+1] = sum exp(wf[i,:] - rowmax)
//   loss       += (logsumexp(logits_i) - logits_i[label]) / B
// ---------------------------------------------------------------------------
__global__ __launch_bounds__(256) void row_stats(const float* __restrict__ wf,
                                                 const int* __restrict__ labels,
                                                 float* __restrict__ stats,
                                                 float* __restrict__ loss) {
  const int i = blockIdx.x;
  const int t = threadIdx.x;
  const float* __restrict__ row = wf + (size_t)i * C;
  const int lab = labels[i];

  const float wl = row[lab];
  const float cl = fminf(fmaxf(wl, -1.0f + EPS), 1.0f - EPS);
  const float lv = SCALE_S * cosf(acosf(cl) + MARGIN);   // margined label logit

  float mp = -INFINITY, sp = 0.0f;   // prediction softmax stats
  float ml = -INFINITY, sl = 0.0f;   // loss logsumexp stats
  for (int j = t; j < C; j += 256) {
    const float v = row[j];
    float m2 = fmaxf(mp, v);
    sp = sp * __expf(mp - m2) + __expf(v - m2);
    mp = m2;
    const float w = (j == lab) ? lv : SCALE_S * v;
    float m3 = fmaxf(ml, w);
    sl = sl * __expf(ml - m3) + __expf(w - m3);
    ml = m3;
  }

  __shared__ float smp[256], ssp[256], sml[256], ssl[256];
  smp[t] = mp; ssp[t] = sp; sml[t] = ml; ssl[t] = sl;
  __syncthreads();
  for (int s = 128; s > 0; s >>= 1) {
    if (t < s) {
      float m1 = smp[t], s1 = ssp[t], m2 = smp[t + s], s2 = ssp[t + s];
      float m = fmaxf(m1, m2);
      smp[t] = m;
      ssp[t] = s1 * __expf(m1 - m) + s2 * __expf(m2 - m);
      m1 = sml[t]; s1 = ssl[t]; m2 = sml[t + s]; s2 = ssl[t + s];
      m = fmaxf(m1, m2);
      sml[t] = m;
      ssl[t] = s1 * __expf(m1 - m) + s2 * __expf(m2 - m);
    }
    __syncthreads();
  }
  if (t == 0) {
    stats[2 * i]     = smp[0];
    stats[2 * i + 1] = ssp[0];
    const float lse = sml[0] + __logf(ssl[0]);
    atomicAdd(loss, (lse - lv) * (1.0f / (float)B));   // -logp[label]/B
  }
}

// ---------------------------------------------------------------------------
// In-place normalize: prediction = exp(wf - rowmax) / rowsum
// ---------------------------------------------------------------------------
__global__ __launch_bounds__(256) void softmax_write(float* __restrict__ wf,
                                                     const float* __restrict__ stats) {
  const int i = blockIdx.y;
  const int j = blockIdx.x * 256 + threadIdx.x;
  if (j >= C) return;
  const float m    = stats[2 * i];
  const float rinv = 1.0f / stats[2 * i + 1];
  const size_t idx = (size_t)i * C + j;
  wf[idx] = __expf(wf[idx] - m) * rinv;
}

// ---------------------------------------------------------------------------
extern "C" void kernel_launch(void* const* d_in, const int* in_sizes, int n_in,
                              void* d_out, int out_size, void* d_ws, size_t ws_size,
                              hipStream_t stream) {
  (void)in_sizes; (void)n_in; (void)out_size; (void)ws_size;

  const float* x      = (const float*)d_in[0];   // [B, D]
  const float* W      = (const float*)d_in[1];   // [C, D]
  const float* bias   = (const float*)d_in[2];   // [C]
  const int*   labels = (const int*)d_in[3];     // [B]

  float* out  = (float*)d_out;
  float* wf   = out;                              // [B*C] prediction region (also raw logits)
  float* loss = out + (size_t)B * C;              // scalar slot

  float* e     = (float*)d_ws;                    // [B*D]
  float* Wn    = e + (size_t)B * D;               // [C*D]
  float* stats = Wn + (size_t)C * D;              // [2*B]

  l2norm_rows<<<B, 128, 0, stream>>>(x, e);
  l2norm_rows<<<C, 128, 0, stream>>>(W, Wn);

  dim3 gg((C + 63) / 64, B / 32);
  gemm_wmma_f32<<<gg, 256, 0, stream>>>(e, Wn, bias, wf);

  zero_loss<<<1, 1, 0, stream>>>(loss);
  row_stats<<<B, 256, 0, stream>>>(wf, labels, stats, loss);

  dim3 gs((C + 255) / 256, B);
  softmax_write<<<gs, 256, 0, stream>>>(wf, stats);
}